// ArmModel_65919158059445
// MI455X (gfx1250) — compile-verified
//
#include <hip/hip_runtime.h>
#include <math.h>
#include <stdint.h>

// ---------------------------------------------------------------------------
// 7-DOF arm RK4 step, one thread per batch element (wave32, gfx1250).
// - Shared model constants staged to LDS via CDNA5 async-to-LDS (ASYNCcnt),
//   derived once per block, broadcast-read by all lanes (no bank conflicts).
// - Mass matrix exploits dq=0,g=0 structure (V=0): unit-accel columns reuse
//   the per-stage adjoint cache; SPD solve via fully unrolled Cholesky.
// - f32 scalar FMA path: every matrix op is per-element 3x3/6x6/7x7 with a
//   sequential joint recurrence -> no wave-shared GEMM exists for WMMA.
// - Outputs written with non-temporal stores (write-once streams).
// ---------------------------------------------------------------------------

#define NJ 7
#define TPB 64

struct ArmConsts {
  float A[NJ][6];        // normalized screw axes
  float Gd[NJ][6];       // spatial inertia diagonals
  float Rinv[NJ][9];     // M_i^{-1} rotation (R^T)
  float tinv[NJ][3];     // M_i^{-1} translation (-R^T p)
  float Mr[NJ + 1][9];   // M_i rotation
  float Mp[NJ + 1][3];   // M_i translation
  float Rend[9];         // Ad(M_7^{-1}) rotation block
  float Lend[9];         // Ad(M_7^{-1}) lower-left block
  float grav[3];
};

__device__ __forceinline__ void mv3(const float* R, const float* x, float* y) {
  y[0] = fmaf(R[0], x[0], fmaf(R[1], x[1], R[2] * x[2]));
  y[1] = fmaf(R[3], x[0], fmaf(R[4], x[1], R[5] * x[2]));
  y[2] = fmaf(R[6], x[0], fmaf(R[7], x[1], R[8] * x[2]));
}
__device__ __forceinline__ void mtv3(const float* R, const float* x, float* y) {
  y[0] = fmaf(R[0], x[0], fmaf(R[3], x[1], R[6] * x[2]));
  y[1] = fmaf(R[1], x[0], fmaf(R[4], x[1], R[7] * x[2]));
  y[2] = fmaf(R[2], x[0], fmaf(R[5], x[1], R[8] * x[2]));
}
__device__ __forceinline__ void mm3(const float* A, const float* B, float* C) {
#pragma unroll
  for (int r = 0; r < 3; ++r)
#pragma unroll
    for (int c = 0; c < 3; ++c)
      C[3 * r + c] = fmaf(A[3 * r + 0], B[c],
                     fmaf(A[3 * r + 1], B[3 + c], A[3 * r + 2] * B[6 + c]));
}
__device__ __forceinline__ void cross3(const float* a, const float* b, float* c) {
  c[0] = fmaf(a[1], b[2], -a[2] * b[1]);
  c[1] = fmaf(a[2], b[0], -a[0] * b[2]);
  c[2] = fmaf(a[0], b[1], -a[1] * b[0]);
}
__device__ __forceinline__ float dot6(const float* a, const float* b) {
  return fmaf(a[0], b[0], fmaf(a[1], b[1], fmaf(a[2], b[2],
         fmaf(a[3], b[3], fmaf(a[4], b[4], a[5] * b[5])))));
}

// Rodrigues: rotation Re and translation pe of exp([A]*theta), |w|=1.
__device__ __forceinline__ void exp_se3_Rp(const float* Aik, float th, float* Re, float* pe) {
  float s, c;
  sincosf(th, &s, &c);
  const float wx = Aik[0], wy = Aik[1], wz = Aik[2];
  const float vx = Aik[3], vy = Aik[4], vz = Aik[5];
  const float o = 1.0f - c;
  Re[0] = fmaf(o * wx, wx, c);       Re[1] = fmaf(o * wx, wy, -s * wz); Re[2] = fmaf(o * wx, wz, s * wy);
  Re[3] = fmaf(o * wx, wy, s * wz);  Re[4] = fmaf(o * wy, wy, c);       Re[5] = fmaf(o * wy, wz, -s * wx);
  Re[6] = fmaf(o * wx, wz, -s * wy); Re[7] = fmaf(o * wy, wz, s * wx);  Re[8] = fmaf(o * wz, wz, c);
  const float wv = fmaf(wx, vx, fmaf(wy, vy, wz * vz));
  float w3[3] = {wx, wy, wz}, v3[3] = {vx, vy, vz}, cw[3];
  cross3(w3, v3, cw);
  const float ts = (th - s) * wv;
  pe[0] = fmaf(s, vx, fmaf(o, cw[0], ts * wx));
  pe[1] = fmaf(s, vy, fmaf(o, cw[1], ts * wy));
  pe[2] = fmaf(s, vz, fmaf(o, cw[2], ts * wz));
}

// y = Ad * x  with Ad = [[R,0],[L,R]]
__device__ __forceinline__ void adApply(const float* R, const float* L, const float* x, float* y) {
  float a[3], b[3], c[3];
  mv3(R, x, a); mv3(L, x, b); mv3(R, x + 3, c);
  y[0] = a[0]; y[1] = a[1]; y[2] = a[2];
  y[3] = b[0] + c[0]; y[4] = b[1] + c[1]; y[5] = b[2] + c[2];
}
// F = Ad^T * F (in place)
__device__ __forceinline__ void adTApply(const float* R, const float* L, float* F) {
  float a[3], b[3], c[3];
  mtv3(R, F, a); mtv3(L, F + 3, b); mtv3(R, F + 3, c);
  F[0] = a[0] + b[0]; F[1] = a[1] + b[1]; F[2] = a[2] + b[2];
  F[3] = c[0]; F[4] = c[1]; F[5] = c[2];
}

// Forward dynamics: acc = M(q)^{-1} (tau - bias(q,dq))
__device__ __forceinline__ void qacc7(const ArmConsts& K, const float* q, const float* dq,
                                      const float* tauv, float* acc) {
  float AdR[NJ][9], AdL[NJ][9];
  float V[NJ][6], Vd[NJ][6];
  // ---- forward pass: link adjoints, twists, accelerations ----
  {
    float Vc[6] = {0, 0, 0, 0, 0, 0};
    float Vdc[6] = {0, 0, 0, -K.grav[0], -K.grav[1], -K.grav[2]};
#pragma unroll
    for (int i = 0; i < NJ; ++i) {
      float Re[9], pe[3], pT[3];
      exp_se3_Rp(&K.A[i][0], -q[i], Re, pe);
      mm3(Re, &K.Rinv[i][0], &AdR[i][0]);
      mv3(Re, &K.tinv[i][0], pT);
      pT[0] += pe[0]; pT[1] += pe[1]; pT[2] += pe[2];
#pragma unroll
      for (int c = 0; c < 3; ++c) {  // L = hat(pT) * R
        AdL[i][c]     = fmaf(-pT[2], AdR[i][3 + c],  pT[1] * AdR[i][6 + c]);
        AdL[i][3 + c] = fmaf( pT[2], AdR[i][c],     -pT[0] * AdR[i][6 + c]);
        AdL[i][6 + c] = fmaf(-pT[1], AdR[i][c],      pT[0] * AdR[i][3 + c]);
      }
      float nV[6], nVd[6];
      adApply(&AdR[i][0], &AdL[i][0], Vc, nV);
#pragma unroll
      for (int k = 0; k < 6; ++k) nV[k] = fmaf(K.A[i][k], dq[i], nV[k]);
      adApply(&AdR[i][0], &AdL[i][0], Vdc, nVd);
      float c1[3], c2[3], c3[3];
      cross3(nV, &K.A[i][0], c1);      // w x Aw
      cross3(nV + 3, &K.A[i][0], c2);  // v x Aw
      cross3(nV, &K.A[i][3], c3);      // w x Av
#pragma unroll
      for (int k = 0; k < 3; ++k) {
        nVd[k]     = fmaf(c1[k], dq[i], nVd[k]);
        nVd[3 + k] = fmaf(c2[k] + c3[k], dq[i], nVd[3 + k]);
      }
#pragma unroll
      for (int k = 0; k < 6; ++k) { V[i][k] = nV[k]; Vd[i][k] = nVd[k]; Vc[k] = nV[k]; Vdc[k] = nVd[k]; }
    }
  }
  // ---- backward pass: bias torques ----
  float bias[NJ];
  {
    float F[6] = {0, 0, 0, 0, 0, 0};
#pragma unroll
    for (int i = NJ - 1; i >= 0; --i) {
      if (i == NJ - 1) adTApply(K.Rend, K.Lend, F);
      else             adTApply(&AdR[i + 1][0], &AdL[i + 1][0], F);
      float Gv[6];
#pragma unroll
      for (int k = 0; k < 6; ++k) {
        Gv[k] = K.Gd[i][k] * V[i][k];
        F[k]  = fmaf(K.Gd[i][k], Vd[i][k], F[k]);
      }
      float cA[3], cB[3], cC[3];
      cross3(&V[i][0], Gv, cA);
      cross3(&V[i][3], Gv + 3, cB);
      cross3(&V[i][0], Gv + 3, cC);
#pragma unroll
      for (int k = 0; k < 3; ++k) { F[k] += cA[k] + cB[k]; F[3 + k] += cC[k]; }
      bias[i] = dot6(F, &K.A[i][0]);
    }
  }
  // ---- mass matrix: unit-accel columns reuse AdT; dq=0,g=0 => V=0 ----
  float Mm[NJ][NJ];
#pragma unroll
  for (int j = 0; j < NJ; ++j) {
    float U[NJ][6];
#pragma unroll
    for (int k = 0; k < 6; ++k) U[j][k] = K.A[j][k];
#pragma unroll
    for (int i = j + 1; i < NJ; ++i) adApply(&AdR[i][0], &AdL[i][0], &U[i - 1][0], &U[i][0]);
    float F2[6] = {0, 0, 0, 0, 0, 0};
#pragma unroll
    for (int i = NJ - 1; i >= j; --i) {
      if (i == NJ - 1) adTApply(K.Rend, K.Lend, F2);
      else             adTApply(&AdR[i + 1][0], &AdL[i + 1][0], F2);
#pragma unroll
      for (int k = 0; k < 6; ++k) F2[k] = fmaf(K.Gd[i][k], U[i][k], F2[k]);
      const float d = dot6(F2, &K.A[i][0]);
      Mm[i][j] = d;
      Mm[j][i] = d;  // symmetric fill
    }
  }
  // ---- Cholesky solve (SPD) ----
  float rhs[NJ], diagInv[NJ], yv[NJ];
#pragma unroll
  for (int i = 0; i < NJ; ++i) rhs[i] = tauv[i] - bias[i];
#pragma unroll
  for (int kk = 0; kk < NJ; ++kk) {
    float s = Mm[kk][kk];
#pragma unroll
    for (int j = 0; j < kk; ++j) s = fmaf(-Mm[kk][j], Mm[kk][j], s);
    const float lkk = sqrtf(s);
    const float dinv = 1.0f / lkk;
    diagInv[kk] = dinv;
#pragma unroll
    for (int i = kk + 1; i < NJ; ++i) {
      float t = Mm[i][kk];
#pragma unroll
      for (int j = 0; j < kk; ++j) t = fmaf(-Mm[i][j], Mm[kk][j], t);
      Mm[i][kk] = t * dinv;
    }
  }
#pragma unroll
  for (int i = 0; i < NJ; ++i) {
    float t = rhs[i];
#pragma unroll
    for (int j = 0; j < i; ++j) t = fmaf(-Mm[i][j], yv[j], t);
    yv[i] = t * diagInv[i];
  }
#pragma unroll
  for (int i = NJ - 1; i >= 0; --i) {
    float t = yv[i];
#pragma unroll
    for (int j = i + 1; j < NJ; ++j) t = fmaf(-Mm[j][i], acc[j], t);
    acc[i] = t * diagInv[i];
  }
}

// CDNA5 async global->LDS staging (ASYNCcnt path), one dword per lane.
__device__ __forceinline__ void stage_async(float* dst, const float* src, int ndw, int tid) {
  for (int i = tid; i < ndw; i += TPB) {
    const unsigned lds = (unsigned)(uintptr_t)(dst + i);  // low 32 bits = LDS offset
    const unsigned off = (unsigned)(i * 4);
    asm volatile("global_load_async_to_lds_b32 %0, %1, %2"
                 :: "v"(lds), "v"(off), "s"(src) : "memory");
  }
}

__global__ void __launch_bounds__(TPB)
arm_rk4_kernel(const float* __restrict__ state, const float* __restrict__ action,
               const float* __restrict__ rawM, const float* __restrict__ rawA,
               const float* __restrict__ rawG, const float* __restrict__ rawGrav,
               float* __restrict__ outState, float* __restrict__ outXY, int batch) {
  __shared__ float raw[201];   // [0,128)=_M  [128,170)=_A  [170,198)=_G  [198,201)=gravity
  __shared__ ArmConsts K;
  const int tid = threadIdx.x;
  const int b = blockIdx.x * TPB + tid;

  if (b < batch) {  // gfx1250 global_prefetch_b8
    __builtin_prefetch(state + (size_t)b * 14, 0, 0);
    __builtin_prefetch(action + (size_t)b * 7, 0, 0);
  }

  stage_async(raw,       rawM,    128, tid);
  stage_async(raw + 128, rawA,    42,  tid);
  stage_async(raw + 170, rawG,    28,  tid);
  stage_async(raw + 198, rawGrav, 3,   tid);
  asm volatile("s_wait_asynccnt 0" ::: "memory");
  __syncthreads();

  // ---- derive shared constants once per block (threads 0..7) ----
  if (tid < NJ + 1) {
    const float* m = raw + tid * 16;
    float a1v[3] = {m[0], m[4], m[8]};
    float a2v[3] = {m[1], m[5], m[9]};
    float pv[3]  = {m[3], m[7], m[11]};
    float n1 = 1.0f / sqrtf(a1v[0] * a1v[0] + a1v[1] * a1v[1] + a1v[2] * a1v[2]);
    float b1[3] = {a1v[0] * n1, a1v[1] * n1, a1v[2] * n1};
    float dp = a2v[0] * b1[0] + a2v[1] * b1[1] + a2v[2] * b1[2];
    float a2o[3] = {a2v[0] - dp * b1[0], a2v[1] - dp * b1[1], a2v[2] - dp * b1[2]};
    float n2 = 1.0f / sqrtf(a2o[0] * a2o[0] + a2o[1] * a2o[1] + a2o[2] * a2o[2]);
    float b2[3] = {a2o[0] * n2, a2o[1] * n2, a2o[2] * n2};
    float b3[3]; cross3(b1, b2, b3);
    float R[9] = {b1[0], b2[0], b3[0], b1[1], b2[1], b3[1], b1[2], b2[2], b3[2]};
#pragma unroll
    for (int k = 0; k < 9; ++k) K.Mr[tid][k] = R[k];
#pragma unroll
    for (int k = 0; k < 3; ++k) K.Mp[tid][k] = pv[k];
    float Rt[9] = {R[0], R[3], R[6], R[1], R[4], R[7], R[2], R[5], R[8]};
    float tin[3]; mv3(Rt, pv, tin);
    tin[0] = -tin[0]; tin[1] = -tin[1]; tin[2] = -tin[2];
    if (tid < NJ) {
#pragma unroll
      for (int k = 0; k < 9; ++k) K.Rinv[tid][k] = Rt[k];
#pragma unroll
      for (int k = 0; k < 3; ++k) K.tinv[tid][k] = tin[k];
    } else {
#pragma unroll
      for (int k = 0; k < 9; ++k) K.Rend[k] = Rt[k];
#pragma unroll
      for (int c = 0; c < 3; ++c) {  // Lend = hat(tin) * Rt
        K.Lend[c]     = fmaf(-tin[2], Rt[3 + c],  tin[1] * Rt[6 + c]);
        K.Lend[3 + c] = fmaf( tin[2], Rt[c],     -tin[0] * Rt[6 + c]);
        K.Lend[6 + c] = fmaf(-tin[1], Rt[c],      tin[0] * Rt[3 + c]);
      }
    }
  }
  if (tid < NJ) {
    const float* av = raw + 128 + tid * 6;
    float nw = 1.0f / sqrtf(av[0] * av[0] + av[1] * av[1] + av[2] * av[2]);
    K.A[tid][0] = av[0] * nw; K.A[tid][1] = av[1] * nw; K.A[tid][2] = av[2] * nw;
    K.A[tid][3] = av[3]; K.A[tid][4] = av[4]; K.A[tid][5] = av[5];
    const float* gv = raw + 170 + tid * 4;
    K.Gd[tid][0] = fabsf(gv[0]); K.Gd[tid][1] = fabsf(gv[1]); K.Gd[tid][2] = fabsf(gv[2]);
    K.Gd[tid][3] = fabsf(gv[3]); K.Gd[tid][4] = fabsf(gv[3]); K.Gd[tid][5] = fabsf(gv[3]);
  }
  if (tid == 0) { K.grav[0] = raw[198]; K.grav[1] = raw[199]; K.grav[2] = raw[200]; }
  __syncthreads();

  if (b >= batch) return;

  // ---- per-element RK4 (inputs read once: non-temporal) ----
  float q[NJ], dq[NJ], tau[NJ];
#pragma unroll
  for (int i = 0; i < NJ; ++i) {
    q[i]   = __builtin_nontemporal_load(state + (size_t)b * 14 + i);
    dq[i]  = __builtin_nontemporal_load(state + (size_t)b * 14 + 7 + i);
    tau[i] = __builtin_nontemporal_load(action + (size_t)b * 7 + i) * 50.0f;  // ACTION_RANGE
  }
  const float h = 0.1f;  // TIMESTEP
  float a1[NJ], a2[NJ], a3[NJ], a4[NJ];
  float q2[NJ], dq2[NJ], q3[NJ], dq3[NJ], q4[NJ], dq4[NJ];

  qacc7(K, q, dq, tau, a1);
#pragma unroll
  for (int i = 0; i < NJ; ++i) { q2[i] = fmaf(0.5f * h, dq[i], q[i]);  dq2[i] = fmaf(0.5f * h, a1[i], dq[i]); }
  qacc7(K, q2, dq2, tau, a2);
#pragma unroll
  for (int i = 0; i < NJ; ++i) { q3[i] = fmaf(0.5f * h, dq2[i], q[i]); dq3[i] = fmaf(0.5f * h, a2[i], dq[i]); }
  qacc7(K, q3, dq3, tau, a3);
#pragma unroll
  for (int i = 0; i < NJ; ++i) { q4[i] = fmaf(h, dq3[i], q[i]);        dq4[i] = fmaf(h, a3[i], dq[i]); }
  qacc7(K, q4, dq4, tau, a4);

  const float PI = 3.14159265358979323846f;
  const float h6 = h / 6.0f;
  float qn[NJ], dqn[NJ];
#pragma unroll
  for (int i = 0; i < NJ; ++i) {
    float qsum  = dq[i] + 2.0f * dq2[i] + 2.0f * dq3[i] + dq4[i];
    float dqsum = a1[i] + 2.0f * a2[i] + 2.0f * a3[i] + a4[i];
    float qv = fmaf(h6, qsum, q[i]);
    float x = fmodf(qv + PI, 2.0f * PI);
    if (x < 0.0f) x += 2.0f * PI;
    qn[i]  = x - PI;
    float dv = fmaf(h6, dqsum, dq[i]);
    dqn[i] = fminf(fmaxf(dv, -20.0f), 20.0f);  // MAX_VELOCITY
    __builtin_nontemporal_store(qn[i],  outState + (size_t)b * 14 + i);
    __builtin_nontemporal_store(dqn[i], outState + (size_t)b * 14 + 7 + i);
  }

  // ---- FK: end-effector (x,y) ----
  float Rw[9] = {1, 0, 0, 0, 1, 0, 0, 0, 1};
  float pw[3] = {0, 0, 0};
#pragma unroll
  for (int i = 0; i < NJ; ++i) {
    float R1[9], p1[3];
    mm3(Rw, &K.Mr[i][0], R1);
    mv3(Rw, &K.Mp[i][0], p1);
    p1[0] += pw[0]; p1[1] += pw[1]; p1[2] += pw[2];
    float Re[9], pe[3];
    exp_se3_Rp(&K.A[i][0], qn[i], Re, pe);
    mm3(R1, Re, Rw);
    mv3(R1, pe, pw);
    pw[0] += p1[0]; pw[1] += p1[1]; pw[2] += p1[2];
  }
  float pf[3];
  mv3(Rw, &K.Mp[NJ][0], pf);
  __builtin_nontemporal_store(pf[0] + pw[0], outXY + (size_t)b * 2 + 0);
  __builtin_nontemporal_store(pf[1] + pw[1], outXY + (size_t)b * 2 + 1);
}

extern "C" void kernel_launch(void* const* d_in, const int* in_sizes, int n_in,
                              void* d_out, int out_size, void* d_ws, size_t ws_size,
                              hipStream_t stream) {
  (void)n_in; (void)out_size; (void)d_ws; (void)ws_size;
  const float* state  = (const float*)d_in[0];
  const float* action = (const float*)d_in[1];
  const float* rawM   = (const float*)d_in[2];
  const float* rawA   = (const float*)d_in[3];
  const float* rawG   = (const float*)d_in[4];
  const float* rawGv  = (const float*)d_in[5];
  const int batch = in_sizes[0] / 14;
  float* out = (float*)d_out;
  float* outXY = out + (size_t)batch * 14;
  const int blocks = (batch + TPB - 1) / TPB;
  arm_rk4_kernel<<<blocks, TPB, 0, stream>>>(state, action, rawM, rawA, rawG, rawGv,
                                             out, outXY, batch);
}